// InterpBaselineEncoder_71786083385356
// MI455X (gfx1250) — compile-verified
//
#include <hip/hip_runtime.h>
#include <math.h>

// Problem constants (match reference)
#define B_   8
#define NC_  2048
#define H_   128
#define W_   128
#define NT_  4096
#define EPS_ 1e-6f

typedef __attribute__((ext_vector_type(2))) float v2f;
typedef __attribute__((ext_vector_type(8))) float v8f;
typedef __attribute__((ext_vector_type(4))) int   v4i;

#define AS1 __attribute__((address_space(1)))
#define AS3 __attribute__((address_space(3)))

// ---------------------------------------------------------------------------
// Kernel 0: per-batch means of yc_off_grid (fill for off-grid) and yc_on_grid
// (fill for on-grid).  16 blocks: blockIdx.x = b*2 + which.
// means[0..7] = mean_off[b], means[8..15] = mean_on[b]
// ---------------------------------------------------------------------------
__global__ __launch_bounds__(256) void InterpMeans_kernel(
    const float* __restrict__ yc_off, const float* __restrict__ yc_on,
    float* __restrict__ means) {
  const int b     = blockIdx.x >> 1;
  const int which = blockIdx.x & 1;
  const float* src = which ? (yc_on + (size_t)b * H_ * W_)
                           : (yc_off + (size_t)b * NC_);
  const int n = which ? (H_ * W_) : NC_;
  float s = 0.f;
  for (int i = threadIdx.x; i < n; i += 256) s += src[i];
  __shared__ float red[8];
  #pragma unroll
  for (int off = 16; off > 0; off >>= 1) s += __shfl_down(s, off, 32);
  if ((threadIdx.x & 31) == 0) red[threadIdx.x >> 5] = s;
  __syncthreads();
  if (threadIdx.x == 0) {
    float tot = 0.f;
    #pragma unroll
    for (int i = 0; i < 8; ++i) tot += red[i];
    means[which * 8 + b] = tot / (float)n;
  }
}

// ---------------------------------------------------------------------------
// Kernel 1: fused off-grid 3-NN (WMMA f32 16x16x4 distance tiles) + barycentric
// interp + on-grid bilinear + 2->1 mix.
// One wave of 32 lanes handles 16 queries.  2048 waves = 256 blocks x 256 thr.
// All 8 waves of a block share one batch -> stage that batch's 2048 points
// (16 KB) into LDS once via async global->LDS (ASYNCcnt path), then the
// 128-tile WMMA loop reads broadcast ds_load_b64 instead of vmem.
//
// WMMA mapping:  D[m][n] = |p_m|^2 - 2 q_n . p_m   (|q|^2 dropped: constant
// per query, does not change nearest-neighbor ordering).
//   A (16x4, points):  row m = [p.x, p.y, |p|^2, 0]
//   B (4x16, queries): col n = [-2q.x, -2q.y, 1, 0]
//   D layout: lane = query n (mod 16); VGPR v = point m=v (lanes0-15) or
//   m=v+8 (lanes16-31)  ->  per-lane top-3 over 8 values, branch-free.
// ---------------------------------------------------------------------------
__global__ __launch_bounds__(256) void InterpBaselineEncoder_71786083385356_kernel(
    const float* __restrict__ xc_off, const float* __restrict__ yc_off,
    const float* __restrict__ xc_on,  const float* __restrict__ yc_on,
    const float* __restrict__ xt,     const float* __restrict__ mix_w,
    const float* __restrict__ mix_b,  const float* __restrict__ means,
    float* __restrict__ out) {
  __shared__ __align__(16) float spts[NC_ * 2];  // 16 KB: this batch's points

  const int wave  = (blockIdx.x * 256 + (int)threadIdx.x) >> 5; // 0..2047
  const int lane  = threadIdx.x & 31;
  const int b     = wave >> 8;                 // 256 waves per batch
  const int qbase = (wave & 255) << 4;         // 16 queries per wave
  const int l16   = lane & 15;
  const bool hi   = lane >= 16;

  // ---- Stage xc_off[b] into LDS (whole block cooperates; same b) ----------
  const float* gsrc = xc_off + (size_t)b * NC_ * 2;
#if __has_builtin(__builtin_amdgcn_global_load_async_to_lds_b128)
  {
    AS1 v4i* g = (AS1 v4i*)(uintptr_t)gsrc;     // global, 16B chunks
    AS3 v4i* l = (AS3 v4i*)spts;                // LDS, 16B chunks
    const int tid = threadIdx.x;
    #pragma unroll
    for (int k = 0; k < 4; ++k) {               // 256 thr * 4 * 16B = 16 KB
      const int c16 = tid + k * 256;
      __builtin_amdgcn_global_load_async_to_lds_b128(g + c16, l + c16, 0, 0);
    }
#if __has_builtin(__builtin_amdgcn_s_wait_asynccnt)
    __builtin_amdgcn_s_wait_asynccnt(0);
#else
    asm volatile("s_wait_asynccnt 0x0" ::: "memory");
#endif
  }
#else
  {
    const float4* g4 = (const float4*)gsrc;
    float4*       l4 = (float4*)spts;
    for (int k = threadIdx.x; k < (NC_ * 2) / 4; k += 256) l4[k] = g4[k];
  }
#endif
  __syncthreads();

  // ---- Query operand (B matrix) -------------------------------------------
  const int   t  = qbase + l16;
  const float qx = xt[(((size_t)b * NT_ + t) << 1) + 0];
  const float qy = xt[(((size_t)b * NT_ + t) << 1) + 1];

  v2f Bm;
  Bm.x = hi ? 1.0f : (-2.0f * qx);
  Bm.y = hi ? 0.0f : (-2.0f * qy);

  // Per-lane sorted top-3 (ascending distance; ties keep lower index because
  // we scan ascending-index with strict '<', matching lax.top_k).
  float d0 = 1e30f, d1 = 1e30f, d2v = 1e30f;
  int   i0 = 0,     i1 = 0,     i2 = 0;

  for (int tile = 0; tile < NC_ / 16; ++tile) {
    const float px = spts[((tile * 16 + l16) << 1) + 0];
    const float py = spts[((tile * 16 + l16) << 1) + 1];
    v2f Am;
    Am.x = hi ? (px * px + py * py) : px;
    Am.y = hi ? 0.0f : py;
    v8f c = {};
    c = __builtin_amdgcn_wmma_f32_16x16x4_f32(
        /*neg_a=*/false, Am, /*neg_b=*/false, Bm,
        /*c_mod=*/(short)0, c, /*reuse_a=*/false, /*reuse_b=*/false);
    const int base = tile * 16 + (hi ? 8 : 0);
    #pragma unroll
    for (int v = 0; v < 8; ++v) {
      const float d   = c[v];
      const int   idx = base + v;
      const bool lt0 = d < d0, lt1 = d < d1, lt2 = d < d2v;
      const float nd2 = lt1 ? d1 : (lt2 ? d : d2v);
      const int   ni2 = lt1 ? i1 : (lt2 ? idx : i2);
      const float nd1 = lt0 ? d0 : (lt1 ? d : d1);
      const int   ni1 = lt0 ? i0 : (lt1 ? idx : i1);
      d0 = lt0 ? d : d0;  i0 = lt0 ? idx : i0;
      d1 = nd1;           i1 = ni1;
      d2v = nd2;          i2 = ni2;
    }
  }

  // Merge the two sorted half-triples (lanes l and l+16 cover disjoint points
  // of the same query).  Tie-break: smaller index first.
  float a0 = d0, a1 = d1, a2 = d2v;
  int   A0 = i0, A1 = i1, A2 = i2;
  float b0 = __shfl_xor(d0, 16, 32), b1 = __shfl_xor(d1, 16, 32),
        b2 = __shfl_xor(d2v, 16, 32);
  int   B0 = __shfl_xor(i0, 16, 32), B1 = __shfl_xor(i1, 16, 32),
        B2 = __shfl_xor(i2, 16, 32);
  float md[3]; int mi[3];
  #pragma unroll
  for (int k = 0; k < 3; ++k) {
    const bool takeA = (a0 < b0) || ((a0 == b0) && (A0 < B0));
    md[k] = takeA ? a0 : b0;
    mi[k] = takeA ? A0 : B0;
    const float na0 = takeA ? a1 : a0; const int nA0 = takeA ? A1 : A0;
    const float na1 = takeA ? a2 : a1; const int nA1 = takeA ? A2 : A1;
    const float nb0 = takeA ? b0 : b1; const int nB0 = takeA ? B0 : B1;
    const float nb1 = takeA ? b1 : b2; const int nB1 = takeA ? B1 : B2;
    a0 = na0; A0 = nA0; a1 = na1; A1 = nA1;
    b0 = nb0; B0 = nB0; b1 = nb1; B1 = nB1;
  }

  // Tail: only low half stores (both halves computed identical merged top-3).
  // Divergence here is fine — all WMMAs are already done.
  if (!hi) {
    const int j0 = mi[0], j1 = mi[1], j2 = mi[2];
    const float p0x = spts[(j0 << 1)], p0y = spts[(j0 << 1) + 1];
    const float p1x = spts[(j1 << 1)], p1y = spts[(j1 << 1) + 1];
    const float p2x = spts[(j2 << 1)], p2y = spts[(j2 << 1) + 1];
    const float tv0 = yc_off[(size_t)b * NC_ + j0];
    const float tv1 = yc_off[(size_t)b * NC_ + j1];
    const float tv2 = yc_off[(size_t)b * NC_ + j2];
    const float e1x = p1x - p0x, e1y = p1y - p0y;
    const float e2x = p2x - p0x, e2y = p2y - p0y;
    const float rx = qx - p0x, ry = qy - p0y;
    const float det = e1x * e2y - e1y * e2x;
    const float det_safe = (fabsf(det) < EPS_) ? 1.0f : det;
    const float w1 = (rx * e2y - ry * e2x) / det_safe;
    const float w2 = (e1x * ry - e1y * rx) / det_safe;
    const float w0 = 1.0f - w1 - w2;
    const bool inside = (w0 >= -EPS_) && (w1 >= -EPS_) && (w2 >= -EPS_) &&
                        (fabsf(det) >= EPS_);
    const float yoff = inside ? (w0 * tv0 + w1 * tv1 + w2 * tv2) : means[b];

    // On-grid bilinear.  gx[i] = xc_on[b,i,0,0], gy[j] = xc_on[b,0,j,1];
    // uniform linspace(-1,1,128) -> cell index via floor, actual grid values
    // loaded for the t fractions (matches searchsorted result a.e.).
    const float* g = xc_on + (size_t)b * H_ * W_ * 2;
    int ix = (int)floorf((qx + 1.0f) * 63.5f);
    ix = ix < 0 ? 0 : (ix > H_ - 2 ? H_ - 2 : ix);
    int iy = (int)floorf((qy + 1.0f) * 63.5f);
    iy = iy < 0 ? 0 : (iy > W_ - 2 ? W_ - 2 : iy);
    const float gx0 = g[(size_t)ix * W_ * 2];
    const float gx1 = g[(size_t)(ix + 1) * W_ * 2];
    const float gy0 = g[iy * 2 + 1];
    const float gy1 = g[(iy + 1) * 2 + 1];
    const float tx = (qx - gx0) / (gx1 - gx0);
    const float ty = (qy - gy0) / (gy1 - gy0);
    const float* vv = yc_on + (size_t)b * H_ * W_;
    const float v00 = vv[ix * W_ + iy];
    const float v01 = vv[ix * W_ + iy + 1];
    const float v10 = vv[(ix + 1) * W_ + iy];
    const float v11 = vv[(ix + 1) * W_ + iy + 1];
    const float gxa = g[0], gxb = g[(size_t)(H_ - 1) * W_ * 2];
    const float gya = g[1], gyb = g[(W_ - 1) * 2 + 1];
    const bool ing = (qx >= gxa) && (qx <= gxb) && (qy >= gya) && (qy <= gyb);
    const float yon = ing ? (v00 * (1.f - tx) * (1.f - ty) +
                             v10 * tx * (1.f - ty) +
                             v01 * (1.f - tx) * ty +
                             v11 * tx * ty)
                          : means[8 + b];

    out[(size_t)b * NT_ + t] = yoff * mix_w[0] + yon * mix_w[1] + mix_b[0];
  }
}

extern "C" void kernel_launch(void* const* d_in, const int* in_sizes, int n_in,
                              void* d_out, int out_size, void* d_ws, size_t ws_size,
                              hipStream_t stream) {
  (void)in_sizes; (void)n_in; (void)out_size; (void)ws_size;
  const float* xc_off = (const float*)d_in[0]; // (8,2048,2)
  const float* yc_off = (const float*)d_in[1]; // (8,2048,1)
  const float* xc_on  = (const float*)d_in[2]; // (8,128,128,2)
  const float* yc_on  = (const float*)d_in[3]; // (8,128,128,1)
  const float* xt     = (const float*)d_in[4]; // (8,4096,2)
  const float* mix_w  = (const float*)d_in[5]; // (2,1)
  const float* mix_b  = (const float*)d_in[6]; // (1,)
  float* out   = (float*)d_out;                // (8,4096,1) f32
  float* means = (float*)d_ws;                 // 16 floats

  InterpMeans_kernel<<<16, 256, 0, stream>>>(yc_off, yc_on, means);
  // 8 batches * 256 query-tiles = 2048 waves -> 256 blocks of 8 waves
  InterpBaselineEncoder_71786083385356_kernel<<<256, 256, 0, stream>>>(
      xc_off, yc_off, xc_on, yc_on, xt, mix_w, mix_b, means, out);
}